// DifferentiablePruner_20143396618864
// MI455X (gfx1250) — compile-verified
//
#include <hip/hip_runtime.h>
#include <math.h>

// ---------------------------------------------------------------------------
// DifferentiablePruner forward for MI455X (gfx1250, wave32, WMMA fp32)
//   out[0:E)               = edge_gates
//   out[E : E+512*512)     = wg1
//   out[E+512*512 : ...)   = wg2
// Block = 128 threads (4 wave32). Each block: 32 edges (2 M-tiles),
// each wave owns one 16-wide slice of H=64 and 2 accumulators.
// ---------------------------------------------------------------------------

typedef __attribute__((ext_vector_type(2))) float v2f;
typedef __attribute__((ext_vector_type(8))) float v8f;

#define DIMF   256          // D
#define KDIM   512          // 2*D  (GEMM K)
#define HDIM   64           // hidden width (GEMM N)
#define PITCH  516          // LDS pitch (floats): bank stride 4 -> conflict free
#define TILE_M 32           // edges per block
#define BETA_INV 10.0f      // 1/0.1

__device__ __forceinline__ float concrete_gate_f(float la, float u) {
    float z = (la + logf(u) - log1pf(-u)) * BETA_INV;
    return 1.0f / (1.0f + expf(-z));
}

// XOR-butterfly add via ds_swizzle (group-of-32 mode: offset = xor<<10 | and=0x1f)
template <int XORMASK>
__device__ __forceinline__ float xor_add(float v) {
    int sw = __builtin_amdgcn_ds_swizzle(__float_as_int(v), (XORMASK << 10) | 0x1f);
    return v + __int_as_float(sw);
}

__global__ void __launch_bounds__(128)
edge_mlp_wmma_kernel(const float* __restrict__ x,
                     const long long* __restrict__ edge_index, // [2,E] int64
                     const float* __restrict__ elog_alpha,     // [E]
                     const float* __restrict__ W1,             // [512,64] row-major
                     const float* __restrict__ b1,             // [64]
                     const float* __restrict__ W2,             // [64]
                     const float* __restrict__ b2,             // [1]
                     const float* __restrict__ u_edge,         // [E]
                     float* __restrict__ out,                  // [E]
                     int E)
{
    __shared__ float feat[TILE_M][PITCH];   // 32 edges x 512 features
    __shared__ float simp[4][TILE_M];       // per-wave partial sims
    __shared__ int   rowi[TILE_M], rowj[TILE_M];

    const int tid = threadIdx.x;
    const int e0  = blockIdx.x * TILE_M;

    if (tid < TILE_M) {
        int e = e0 + tid;
        if (e >= E) e = E - 1;              // clamp: EXEC stays full for WMMA
        int ri = (int)edge_index[e];
        int rj = (int)edge_index[(long long)E + e];
        rowi[tid] = ri;
        rowj[tid] = rj;
        __builtin_prefetch(x + (size_t)ri * DIMF, 0, 0);   // global_prefetch_b8
        __builtin_prefetch(x + (size_t)rj * DIMF, 0, 0);
    }
    __syncthreads();

    // ---- Stage 32 x 512 feature tile into LDS (coalesced float4) ----------
    for (int idx = tid; idx < TILE_M * (KDIM / 4); idx += 128) {
        int row = idx >> 7;                 // / (KDIM/4)
        int c4  = idx & 127;
        int col = c4 << 2;
        const float* src = (col < DIMF)
            ? (x + (size_t)rowi[row] * DIMF + col)
            : (x + (size_t)rowj[row] * DIMF + (col - DIMF));
        float4 v = *(const float4*)src;
        *(float4*)&feat[row][col] = v;
    }
    __syncthreads();

    // ---- fp32 WMMA: h[32 x 64] = feat @ W1, per wave: 1 N-tile x 2 M-tiles
    const int wave = tid >> 5;              // N-tile id
    const int lane = tid & 31;
    const int half = lane >> 4;             // 0: K pair (0,1)  1: K pair (2,3)
    const int nl   = lane & 15;             // A row (M) / B,C column (N)
    const int n0   = wave * 16;

    v8f acc0 = {};                          // edges e0 + 0..15
    v8f acc1 = {};                          // edges e0 + 16..31
    const float* w1col = W1 + (n0 + nl);    // column (n0+nl) of W1
    const int koff = 2 * half;

    #pragma unroll 4
    for (int k0 = 0; k0 < KDIM; k0 += 4) {
        const int ka = k0 + koff;
        v2f b;                               // B: 4x16, shared by both M-tiles
        b.x = w1col[(size_t)ka * HDIM];
        b.y = w1col[(size_t)(ka + 1) * HDIM];
        v2f a0, a1;                          // A: 16x4 per M-tile (from LDS)
        a0.x = feat[nl][ka];
        a0.y = feat[nl][ka + 1];
        a1.x = feat[16 + nl][ka];
        a1.y = feat[16 + nl][ka + 1];
        acc0 = __builtin_amdgcn_wmma_f32_16x16x4_f32(
            false, a0, false, b, (short)0, acc0, false, false);
        acc1 = __builtin_amdgcn_wmma_f32_16x16x4_f32(
            false, a1, false, b, (short)0, acc1, false, false);
    }

    // ---- ReLU + W2 projection + 16-lane XOR reduction ---------------------
    // C/D layout: VGPR r -> row r (lanes 0-15), row r+8 (lanes 16-31); N = nl.
    const float b1v = b1[n0 + nl];
    const float w2v = W2[n0 + nl];
    #pragma unroll
    for (int r = 0; r < 8; ++r) {
        float v = acc0[r] + b1v;
        v = (v > 0.0f ? v : 0.0f) * w2v;
        v = xor_add<1>(v); v = xor_add<2>(v); v = xor_add<4>(v); v = xor_add<8>(v);
        if (nl == 0) simp[wave][r + half * 8] = v;

        float w = acc1[r] + b1v;
        w = (w > 0.0f ? w : 0.0f) * w2v;
        w = xor_add<1>(w); w = xor_add<2>(w); w = xor_add<4>(w); w = xor_add<8>(w);
        if (nl == 0) simp[wave][16 + r + half * 8] = w;
    }
    __syncthreads();

    // ---- Combine the 4 N-tiles, bias + concrete gate, store ---------------
    if (tid < TILE_M) {
        int e = e0 + tid;
        if (e < E) {
            float s = simp[0][tid] + simp[1][tid] + simp[2][tid] + simp[3][tid]
                    + b2[0];
            float g = concrete_gate_f(elog_alpha[e], u_edge[e]);
            out[e] = s * g;
        }
    }
}

// ---- Elementwise weight gates (memory-trivial) ----------------------------
__global__ void __launch_bounds__(256)
wgate_kernel(const float* __restrict__ wla1, const float* __restrict__ u1,
             const float* __restrict__ wla2, const float* __restrict__ u2,
             float* __restrict__ out1, float* __restrict__ out2,
             int n1, int n2)
{
    int t = blockIdx.x * blockDim.x + threadIdx.x;
    if (t < n1) out1[t] = concrete_gate_f(wla1[t], u1[t]);
    if (t < n2) out2[t] = concrete_gate_f(wla2[t], u2[t]);
}

extern "C" void kernel_launch(void* const* d_in, const int* in_sizes, int n_in,
                              void* d_out, int out_size, void* d_ws, size_t ws_size,
                              hipStream_t stream)
{
    const float*     x    = (const float*)d_in[0];
    const long long* ei   = (const long long*)d_in[1];   // int64 edge_index
    const float*     ela  = (const float*)d_in[2];
    const float*     W1   = (const float*)d_in[3];
    const float*     b1   = (const float*)d_in[4];
    const float*     W2   = (const float*)d_in[5];
    const float*     b2   = (const float*)d_in[6];
    const float*     wla1 = (const float*)d_in[7];
    const float*     wla2 = (const float*)d_in[8];
    const float*     ue   = (const float*)d_in[9];
    const float*     u1   = (const float*)d_in[10];
    const float*     u2   = (const float*)d_in[11];

    float* out = (float*)d_out;

    const int E  = in_sizes[2];   // edge_log_alpha count
    const int n1 = in_sizes[7];   // 512*512
    const int n2 = in_sizes[8];   // 512*256

    const int ntiles = (E + TILE_M - 1) / TILE_M;
    edge_mlp_wmma_kernel<<<ntiles, 128, 0, stream>>>(
        x, ei, ela, W1, b1, W2, b2, ue, out, E);

    const int nmax = n1 > n2 ? n1 : n2;
    wgate_kernel<<<(nmax + 255) / 256, 256, 0, stream>>>(
        wla1, u1, wla2, u2, out + E, out + E + n1, n1, n2);
}